// GinApp_39702677684365
// MI455X (gfx1250) — compile-verified
//
#include <hip/hip_runtime.h>

// -------- CDNA5 (gfx1250, wave32) fused GNN forward --------
// Pipeline: walk_kernel (one thread per (element,walk)) ->
//           repack_kernel (W -> WMMA fragment-major layout) ->
//           fused_kernel (float4 gathers -> dual WMMA f32 GEMM -> attention).

typedef __attribute__((ext_vector_type(2))) float v2f;
typedef __attribute__((ext_vector_type(8))) float v8f;

#define DEG   32
#define DIM   128
#define HID   128
#define LWALK 5
#define WLEN  6     // L+1 nodes per walk
#define ASTRIDE 260 // A-matrix LDS row stride (dwords): 16B-aligned rows, bank-rotated
#define FSTRIDE 132 // feature LDS row stride (dwords): 16B-aligned rows

// ---------------- deterministic PCG hash RNG ----------------
__device__ __forceinline__ unsigned pcg_hash(unsigned x) {
    x = x * 747796405u + 2891336453u;
    unsigned w = ((x >> ((x >> 28u) + 4u)) ^ x) * 277803737u;
    return (w >> 22u) ^ w;
}

// ---------------- Kernel 1: weighted random walks ----------------
// One thread per (batch element, walk): halves the dependent-gather chain.
// Inverse-CDF sampling over 32 edge weights (== categorical over log w).
__global__ __launch_bounds__(256)
void walk_kernel(const int* __restrict__ neighbors,
                 const float* __restrict__ edge_w,
                 const int* __restrict__ tgt,
                 int B, int* __restrict__ walks /* [2][B][WLEN] */) {
    int tid = blockIdx.x * blockDim.x + threadIdx.x;
    if (tid >= 2 * B) return;
    const int w = (tid >= B) ? 1 : 0;
    const int b = tid - w * B;

    int cur = tgt[b];
    walks[((long)w * B + b) * WLEN + 0] = cur;
    for (int s = 1; s <= LWALK; ++s) {
        long base = (long)cur * DEG;
        float wt[DEG];
        float tot = 0.f;
        #pragma unroll
        for (int i = 0; i < DEG; ++i) { wt[i] = edge_w[base + i]; tot += wt[i]; }
        unsigned h = pcg_hash((unsigned)b * 0x9E3779B9u
                              ^ (unsigned)(w * 0x85EBCA6Bu)
                              ^ (unsigned)(s * 0xC2B2AE35u));
        float r = ((h >> 8) * (1.0f / 16777216.0f)) * tot;
        float acc = 0.f; int idx = DEG - 1;
        #pragma unroll
        for (int i = 0; i < DEG; ++i) {
            acc += wt[i];
            if (idx == DEG - 1 && acc > r) idx = i;
        }
        cur = neighbors[base + idx];
        walks[((long)w * B + b) * WLEN + s] = cur;
    }
}

// ---------------- Kernel 2: repack W into WMMA fragment-major layout ----------
// Wpk[ct][k4][lane][0..1] = W[4*k4 + 2*(lane>>4) + j][16*ct + (lane&15)]
// One thread per (ct,k4,lane) = 16384 threads; one b64 store each.
__global__ __launch_bounds__(256)
void repack_kernel(const float* __restrict__ W_app,
                   const float* __restrict__ W_loc,
                   float* __restrict__ PkA, float* __restrict__ PkL) {
    const int tid  = blockIdx.x * blockDim.x + threadIdx.x;   // 0..16383
    const int lane = tid & 31;
    const int k4   = (tid >> 5) & 63;
    const int ct   = tid >> 11;
    const int k    = 4 * k4 + 2 * (lane >> 4);
    const int col  = ct * 16 + (lane & 15);
    float2 a, l;
    a.x = W_app[(long)k * HID + col];
    a.y = W_app[(long)(k + 1) * HID + col];
    l.x = W_loc[(long)k * HID + col];
    l.y = W_loc[(long)(k + 1) * HID + col];
    ((float2*)PkA)[tid] = a;
    ((float2*)PkL)[tid] = l;
}

// ---------------- Kernel 3: fused gather + GEMM + attention ----------------
__global__ __launch_bounds__(256)
void fused_kernel(const float* __restrict__ x_app,
                  const float* __restrict__ x_loc,
                  const float* __restrict__ x_time,
                  const int* __restrict__ tgt,
                  const int* __restrict__ walks,
                  int B,
                  const float* __restrict__ PkA, const float* __restrict__ b_app,
                  const float* __restrict__ PkL, const float* __restrict__ b_loc,
                  const float* __restrict__ W_att, const float* __restrict__ b_att,
                  float* __restrict__ out) {
    __shared__ float Aapp[16][ASTRIDE];
    __shared__ float Aloc[16][ASTRIDE];
    __shared__ float Fapp[16][FSTRIDE];
    __shared__ float Floc[16][FSTRIDE];
    __shared__ float Ttim[16][FSTRIDE];
    __shared__ float Sc[16][4];

    const int tid = threadIdx.x;
    const int r   = tid >> 4;     // 0..15 : row within tile
    const int l16 = tid & 15;     // 0..15 : lane group within row
    const long b  = (long)blockIdx.x * 16 + r;
    const int  t  = tgt[b];

    // ---- Phase 1: float4 gathers (2 x b128 per array per thread) ----
    int an[WLEN], ln[WLEN];
    {
        const int* aw = walks + ((long)0 * B + b) * WLEN;
        const int* lw = walks + ((long)1 * B + b) * WLEN;
        #pragma unroll
        for (int s = 0; s < WLEN; ++s) { an[s] = aw[s]; ln[s] = lw[s]; }
    }
    #pragma unroll
    for (int c = 0; c < 2; ++c) {
        const int f = 4 * l16 + 64 * c;                 // element offset
        float4 ta = *(const float4*)&x_app [(long)t * DIM + f];
        float4 tl = *(const float4*)&x_loc [(long)t * DIM + f];
        float4 tt = *(const float4*)&x_time[(long)t * HID + f];
        float4 sa = make_float4(0.f, 0.f, 0.f, 0.f);
        float4 sl = make_float4(0.f, 0.f, 0.f, 0.f);
        #pragma unroll
        for (int s = 0; s < WLEN; ++s) {
            float4 va = *(const float4*)&x_app[(long)an[s] * DIM + f];
            float4 vl = *(const float4*)&x_loc[(long)ln[s] * DIM + f];
            sa.x += va.x; sa.y += va.y; sa.z += va.z; sa.w += va.w;
            sl.x += vl.x; sl.y += vl.y; sl.z += vl.z; sl.w += vl.w;
        }
        *(float4*)&Aapp[r][f]       = ta;
        *(float4*)&Aloc[r][f]       = tl;
        *(float4*)&Ttim[r][f]       = tt;
        *(float4*)&Aapp[r][DIM + f] = sa;
        *(float4*)&Aloc[r][DIM + f] = sl;
    }
    __syncthreads();

    // ---- Phase 2: dual GEMM via V_WMMA_F32_16X16X4_F32, K=256 ----
    const int wave  = tid >> 5;                  // 0..7 -> column tile
    const int lane  = tid & 31;
    const int mrow  = lane & 15;                 // A row (M) this lane feeds
    const int khalf = lane >> 4;                 // K sub-pair (0: K0/K1, 1: K2/K3)
    const int col   = wave * 16 + (lane & 15);   // output column (N)

    const float2* __restrict__ WA = (const float2*)PkA + ((long)wave * 64) * 32 + lane;
    const float2* __restrict__ WL = (const float2*)PkL + ((long)wave * 64) * 32 + lane;

    const float bias_a = b_app[col];
    const float bias_l = b_loc[col];
    v8f ca, cl;
    #pragma unroll
    for (int i = 0; i < 8; ++i) { ca[i] = bias_a; cl[i] = bias_l; }

    for (int k4 = 0; k4 < 64; ++k4) {
        const int k = 4 * k4 + 2 * khalf;
        v2f aA, aL, bA, bL;
        aA[0] = Aapp[mrow][k];  aA[1] = Aapp[mrow][k + 1];
        aL[0] = Aloc[mrow][k];  aL[1] = Aloc[mrow][k + 1];
        float2 fa = WA[k4 * 32];                 // contiguous b64 per fragment
        float2 fl = WL[k4 * 32];
        bA[0] = fa.x; bA[1] = fa.y;
        bL[0] = fl.x; bL[1] = fl.y;
        ca = __builtin_amdgcn_wmma_f32_16x16x4_f32(false, aA, false, bA,
                                                   (short)0, ca, false, false);
        cl = __builtin_amdgcn_wmma_f32_16x16x4_f32(false, aL, false, bL,
                                                   (short)0, cl, false, false);
    }
    // ReLU + stage to LDS (D layout: VGPR i -> row i (lanes 0-15) / i+8 (16-31))
    #pragma unroll
    for (int i = 0; i < 8; ++i) {
        int rr = (lane < 16) ? i : (i + 8);
        Fapp[rr][col] = fmaxf(ca[i], 0.f);
        Floc[rr][col] = fmaxf(cl[i], 0.f);
    }
    __syncthreads();

    // ---- Phase 3: cross-type attention ----
    // s_k = (t_app . W_att[0:128]) + (feat_k . W_att[128:256]) + b_att
    float pt = 0.f, p0 = 0.f, p1 = 0.f, p2 = 0.f;
    #pragma unroll
    for (int c = 0; c < 2; ++c) {
        const int j = 4 * l16 + 64 * c;
        float4 wt = *(const float4*)&W_att[j];
        float4 wf = *(const float4*)&W_att[DIM + j];
        float4 a  = *(const float4*)&Aapp[r][j];
        float4 f0 = *(const float4*)&Fapp[r][j];
        float4 f1 = *(const float4*)&Floc[r][j];
        float4 f2 = *(const float4*)&Ttim[r][j];
        pt += a.x * wt.x + a.y * wt.y + a.z * wt.z + a.w * wt.w;
        p0 += f0.x * wf.x + f0.y * wf.y + f0.z * wf.z + f0.w * wf.w;
        p1 += f1.x * wf.x + f1.y * wf.y + f1.z * wf.z + f1.w * wf.w;
        p2 += f2.x * wf.x + f2.y * wf.y + f2.z * wf.z + f2.w * wf.w;
    }
    // reduce over the 16-lane group (stays within one half of the wave32)
    #pragma unroll
    for (int off = 8; off >= 1; off >>= 1) {
        pt += __shfl_xor(pt, off, 32);
        p0 += __shfl_xor(p0, off, 32);
        p1 += __shfl_xor(p1, off, 32);
        p2 += __shfl_xor(p2, off, 32);
    }
    if (l16 == 0) {
        float ba = b_att[0];
        float s0 = pt + p0 + ba, s1 = pt + p1 + ba, s2 = pt + p2 + ba;
        float m  = fmaxf(s0, fmaxf(s1, s2));
        float e0 = __expf(s0 - m), e1 = __expf(s1 - m), e2 = __expf(s2 - m);
        float inv = 1.f / (e0 + e1 + e2);
        Sc[r][0] = e0 * inv; Sc[r][1] = e1 * inv; Sc[r][2] = e2 * inv;
    }
    __syncthreads();

    const float s0 = Sc[r][0], s1 = Sc[r][1], s2 = Sc[r][2];
    #pragma unroll
    for (int c = 0; c < 2; ++c) {
        const int j = 4 * l16 + 64 * c;
        float4 f0 = *(const float4*)&Fapp[r][j];
        float4 f1 = *(const float4*)&Floc[r][j];
        float4 f2 = *(const float4*)&Ttim[r][j];
        float4 o;
        o.x = s0 * f0.x + s1 * f1.x + s2 * f2.x;
        o.y = s0 * f0.y + s1 * f1.y + s2 * f2.y;
        o.z = s0 * f0.z + s1 * f1.z + s2 * f2.z;
        o.w = s0 * f0.w + s1 * f1.w + s2 * f2.w;
        *(float4*)&out[b * HID + j] = o;
    }
}

// ---------------- host-side launcher ----------------
extern "C" void kernel_launch(void* const* d_in, const int* in_sizes, int n_in,
                              void* d_out, int out_size, void* d_ws, size_t ws_size,
                              hipStream_t stream) {
    const float* x_app   = (const float*)d_in[0];
    const float* x_loc   = (const float*)d_in[1];
    const float* x_time  = (const float*)d_in[2];
    const int*   nbrs    = (const int*)  d_in[3];
    const float* edge_w  = (const float*)d_in[4];
    const int*   tgt     = (const int*)  d_in[5];
    const float* W_app   = (const float*)d_in[6];
    const float* b_app   = (const float*)d_in[7];
    const float* W_loc   = (const float*)d_in[8];
    const float* b_loc   = (const float*)d_in[9];
    const float* W_att   = (const float*)d_in[10];
    const float* b_att   = (const float*)d_in[11];
    float* out = (float*)d_out;

    const int B = in_sizes[5];            // target_nodes count (16384)

    // workspace layout: walks (2*B*6 ints) | PkA (128KB) | PkL (128KB)
    int* walks = (int*)d_ws;
    size_t walkBytes = (size_t)2 * B * WLEN * sizeof(int);
    float* PkA = (float*)((char*)d_ws + ((walkBytes + 255) & ~(size_t)255));
    float* PkL = PkA + 8 * 64 * 32 * 2;

    walk_kernel<<<(2 * B + 255) / 256, 256, 0, stream>>>(nbrs, edge_w, tgt, B, walks);
    repack_kernel<<<64, 256, 0, stream>>>(W_app, W_loc, PkA, PkL);
    fused_kernel<<<B / 16, 256, 0, stream>>>(x_app, x_loc, x_time, tgt, walks, B,
                                             PkA, b_app, PkL, b_loc,
                                             W_att, b_att, out);
}